// Qu_mutual_17454747091657
// MI455X (gfx1250) — compile-verified
//
#include <hip/hip_runtime.h>
#include <math.h>

typedef __attribute__((ext_vector_type(2))) float v2f;
typedef __attribute__((ext_vector_type(8))) float v8f;

#define LDA 68          // padded LDS row stride (floats) to avoid bank conflicts
#define PLANE (64*LDA)  // one 64x64 padded plane

// ---------------------------------------------------------------------------
// complex helpers
struct cplx { float re, im; };
__device__ __forceinline__ cplx cmul(cplx a, cplx b){
  return { a.re*b.re - a.im*b.im, a.re*b.im + a.im*b.re };
}

// composite CNOT permutation for 12 qubits, 1 layer (pure bit math)
__device__ __forceinline__ unsigned perm12(unsigned p){
  p ^= (p & 1u) << 11;                  // gate (11,0) applied first (reversed order)
  #pragma unroll
  for (int c = 10; c >= 0; --c) {       // gates (c, c+1), c = 10 .. 0
    unsigned bc = 11u - (unsigned)c;    // control bit
    unsigned bt = 10u - (unsigned)c;    // target bit
    p ^= ((p >> bc) & 1u) << bt;
  }
  return p;
}

// ---------------------------------------------------------------------------
// WMMA f32 16x16x4 fragment helpers (wave32)
// A (16x4):  lanes 0-15 row M, VGPR0=K0 VGPR1=K1 ; lanes 16-31: VGPR0=K2 VGPR1=K3
// B (4x16):  lanes 0-15 col N, VGPR0=rowK0 VGPR1=rowK1 ; lanes16-31: K2/K3
// C/D (16x16): VGPR v: lanes0-15 -> M=v, lanes16-31 -> M=v+8 ; N = lane%16
__device__ __forceinline__ v2f fA(const float* S, int m0, int k0, int lr, int hf){
  const float* p = S + (m0 + lr)*LDA + k0 + 2*hf;
  return v2f{p[0], p[1]};
}
__device__ __forceinline__ v2f fB(const float* S, int k0, int c0, int lr, int hf){
  int k = k0 + 2*hf;
  return v2f{ S[k*LDA + c0 + lr], S[(k+1)*LDA + c0 + lr] };
}
// B[k,c] taken from S[c,k]  (transposed read, used for the L-dagger contraction)
__device__ __forceinline__ v2f fBT(const float* S, int k0, int c0, int lr, int hf){
  const float* p = S + (c0 + lr)*LDA + k0 + 2*hf;
  return v2f{p[0], p[1]};
}
__device__ __forceinline__ void storeD(float* S, v8f d, int m0, int c0, int lr, int hf){
  #pragma unroll
  for (int v = 0; v < 8; ++v)
    S[(m0 + v + 8*hf)*LDA + c0 + lr] = d[v];
}

#define WMMA(a,b,c) __builtin_amdgcn_wmma_f32_16x16x4_f32(false,(a),false,(b),(short)0,(c),false,false)

// ---------------------------------------------------------------------------
// Complex sandwich  G = L * M * conj(L)^T  on 64x64 tiles resident in LDS.
// Planes: Lr,Li (input, preserved), Mr,Mi (input; overwritten with G), T1r,T1i scratch.
// Executed by a full 256-thread (8-wave) block; EXEC is all-ones around WMMA.
__device__ __forceinline__ void sandwich_core(const float* Lr, const float* Li,
                                              float* Mr, float* Mi,
                                              float* T1r, float* T1i){
  const int lane = threadIdx.x & 31;
  const int wave = threadIdx.x >> 5;
  const int lr = lane & 15, hf = lane >> 4;

  // ---- step 1: T1 = L * M   (T1r = Lr*Mr - Li*Mi ; T1i = Lr*Mi + Li*Mr)
  #pragma unroll
  for (int t = wave*2; t < wave*2 + 2; ++t){
    int m0 = (t >> 2) * 16, c0 = (t & 3) * 16;
    v8f ar = {}, ai = {};
    for (int k0 = 0; k0 < 64; k0 += 4){
      v2f LaR = fA(Lr, m0, k0, lr, hf);
      v2f LaI = fA(Li, m0, k0, lr, hf);
      v2f MbR = fB(Mr, k0, c0, lr, hf);
      v2f MbI = fB(Mi, k0, c0, lr, hf);
      v2f MbIn = -MbI;                    // f32 WMMA has no A/B neg bits
      ar = WMMA(LaR, MbR, ar);
      ar = WMMA(LaI, MbIn, ar);
      ai = WMMA(LaR, MbI, ai);
      ai = WMMA(LaI, MbR, ai);
    }
    storeD(T1r, ar, m0, c0, lr, hf);
    storeD(T1i, ai, m0, c0, lr, hf);
  }
  __syncthreads();

  // ---- step 2: G = T1 * L^H  with B[k,c] = conj(L[c,k])
  //      Gr = T1r*Lr^T + T1i*Li^T ; Gi = T1i*Lr^T - T1r*Li^T
  #pragma unroll
  for (int t = wave*2; t < wave*2 + 2; ++t){
    int m0 = (t >> 2) * 16, c0 = (t & 3) * 16;
    v8f gr = {}, gi = {};
    for (int k0 = 0; k0 < 64; k0 += 4){
      v2f aR = fA(T1r, m0, k0, lr, hf);
      v2f aI = fA(T1i, m0, k0, lr, hf);
      v2f bR = fBT(Lr, k0, c0, lr, hf);
      v2f bI = fBT(Li, k0, c0, lr, hf);
      v2f bIn = -bI;
      gr = WMMA(aR, bR, gr);
      gr = WMMA(aI, bI, gr);
      gi = WMMA(aI, bR, gi);
      gi = WMMA(aR, bIn, gi);
    }
    storeD(Mr, gr, m0, c0, lr, hf);   // overwrite M planes with G
    storeD(Mi, gi, m0, c0, lr, hf);
  }
  __syncthreads();
}

// ---------------------------------------------------------------------------
// K1: build LA = kron(G0..G5), LB = kron(G6..G11) from weight
__global__ void build_L_kernel(const float* __restrict__ w,
                               float* __restrict__ LAr, float* __restrict__ LAi,
                               float* __restrict__ LBr, float* __restrict__ LBi){
  const float WM = 0.63245553203367586f;   // sqrt(2/5)
  cplx G[12][2][2];
  #pragma unroll
  for (int q = 0; q < 12; ++q){
    float tx = w[q]      * WM * 0.5f;
    float ty = w[12 + q] * WM * 0.5f;
    float tz = w[24 + q] * WM * 0.5f;
    float cx = cosf(tx), sx = sinf(tx);
    float cy = cosf(ty), sy = sinf(ty);
    float cz = cosf(tz), sz = sinf(tz);
    // M = Ry @ Rx
    cplx m00 = { cy*cx,  sy*sx}, m01 = {-sy*cx, -cy*sx};
    cplx m10 = { sy*cx, -cy*sx}, m11 = { cy*cx, -sy*sx};
    cplx e0 = {cz, -sz}, e1 = {cz, sz};   // exp(-i tz/2), exp(+i tz/2)
    G[q][0][0] = cmul(e0, m00); G[q][0][1] = cmul(e0, m01);
    G[q][1][0] = cmul(e1, m10); G[q][1][1] = cmul(e1, m11);
  }
  for (int idx = threadIdx.x; idx < 4096; idx += blockDim.x){
    int r = idx >> 6, c = idx & 63;
    cplx a = {1.f, 0.f}, b = {1.f, 0.f};
    #pragma unroll
    for (int q = 0; q < 6; ++q){
      int rb = (r >> (5 - q)) & 1, cb = (c >> (5 - q)) & 1;
      a = cmul(a, G[q][rb][cb]);        // qubit 0 most significant
      b = cmul(b, G[6 + q][rb][cb]);
    }
    LAr[idx] = a.re; LAi[idx] = a.im;
    LBr[idx] = b.re; LBi[idx] = b.im;
  }
}

// ---------------------------------------------------------------------------
// K2: GX = LA*X*LA^H and GY = LB*Y*LB^H for the 4 combos (8 blocks)
__global__ void __launch_bounds__(256)
phaseA_kernel(const float* __restrict__ A, const float* __restrict__ B,
              const float* __restrict__ LAr, const float* __restrict__ LAi,
              const float* __restrict__ LBr, const float* __restrict__ LBi,
              float* __restrict__ G){
  extern __shared__ float lds[];
  float* Lr  = lds;            float* Li  = lds +   PLANE;
  float* Mr  = lds + 2*PLANE;  float* Mi  = lds + 3*PLANE;
  float* T1r = lds + 4*PLANE;  float* T1i = lds + 5*PLANE;

  int combo = blockIdx.x >> 1, which = blockIdx.x & 1;
  int batch = combo & 1, swp = combo >> 1;
  const float* Xp = (swp ? B : A) + batch*4096;
  const float* Yp = (swp ? A : B) + batch*4096;
  const float* Mg  = which ? Yp  : Xp;
  const float* Lgr = which ? LBr : LAr;
  const float* Lgi = which ? LBi : LAi;

  for (int i = threadIdx.x; i < 4096; i += blockDim.x){
    int r = i >> 6, c = i & 63;
    Lr[r*LDA + c] = Lgr[i];  Li[r*LDA + c] = Lgi[i];
    Mr[r*LDA + c] = Mg[i];   Mi[r*LDA + c] = 0.f;     // real input
  }
  __syncthreads();
  sandwich_core(Lr, Li, Mr, Mi, T1r, T1i);

  float* Go = G + (size_t)(combo*2 + which) * 2 * 4096;
  for (int i = threadIdx.x; i < 4096; i += blockDim.x){
    int r = i >> 6, c = i & 63;
    Go[i]        = Mr[r*LDA + c];
    Go[4096 + i] = Mi[r*LDA + c];
  }
}

// ---------------------------------------------------------------------------
// K3: E[a,a'] = sum_b GX[pa(a,b),pa(a',b)] * GY[pb(a,b),pb(a',b)]  (4 blocks)
__global__ void __launch_bounds__(256)
E_kernel(const float* __restrict__ G, float* __restrict__ E){
  extern __shared__ float sm[];          // gxr|gxi|gyr|gyi : 4*4096 floats
  float* gxr = sm;          float* gxi = sm + 4096;
  float* gyr = sm + 2*4096; float* gyi = sm + 3*4096;
  int combo = blockIdx.x;
  const float* Gc = G + (size_t)combo * 4 * 4096;
  for (int i = threadIdx.x; i < 4096; i += blockDim.x){
    gxr[i] = Gc[i];            gxi[i] = Gc[4096 + i];
    gyr[i] = Gc[2*4096 + i];   gyi[i] = Gc[3*4096 + i];
  }
  __syncthreads();
  for (int idx = threadIdx.x; idx < 4096; idx += blockDim.x){
    int a = idx >> 6, a2 = idx & 63;
    float er = 0.f, ei = 0.f;
    for (int b = 0; b < 64; ++b){
      unsigned m1 = perm12((unsigned)(a  * 64 + b));
      unsigned m2 = perm12((unsigned)(a2 * 64 + b));
      int pa1 = m1 >> 6, pb1 = m1 & 63;
      int pa2 = m2 >> 6, pb2 = m2 & 63;
      float xr = gxr[pa1*64 + pa2], xi = gxi[pa1*64 + pa2];
      float yr = gyr[pb1*64 + pb2], yi = gyi[pb1*64 + pb2];
      er += xr*yr - xi*yi;
      ei += xr*yi + xi*yr;
    }
    E[(size_t)combo*2*4096 + idx]        = er;
    E[(size_t)combo*2*4096 + 4096 + idx] = ei;
  }
}

// ---------------------------------------------------------------------------
// K4: out = LA * E * LA^H -> interleaved complex64 output  (4 blocks)
__global__ void __launch_bounds__(256)
phaseC_kernel(const float* __restrict__ E,
              const float* __restrict__ LAr, const float* __restrict__ LAi,
              float* __restrict__ out){
  extern __shared__ float lds[];
  float* Lr  = lds;            float* Li  = lds +   PLANE;
  float* Mr  = lds + 2*PLANE;  float* Mi  = lds + 3*PLANE;
  float* T1r = lds + 4*PLANE;  float* T1i = lds + 5*PLANE;
  int combo = blockIdx.x;
  const float* Ec = E + (size_t)combo * 2 * 4096;
  for (int i = threadIdx.x; i < 4096; i += blockDim.x){
    int r = i >> 6, c = i & 63;
    Lr[r*LDA + c] = LAr[i];  Li[r*LDA + c] = LAi[i];
    Mr[r*LDA + c] = Ec[i];   Mi[r*LDA + c] = Ec[4096 + i];
  }
  __syncthreads();
  sandwich_core(Lr, Li, Mr, Mi, T1r, T1i);

  float* O = out + (size_t)combo * 2 * 4096;     // complex64 interleaved
  for (int i = threadIdx.x; i < 4096; i += blockDim.x){
    int r = i >> 6, c = i & 63;
    O[2*i]     = Mr[r*LDA + c];
    O[2*i + 1] = Mi[r*LDA + c];
  }
}

// ---------------------------------------------------------------------------
extern "C" void kernel_launch(void* const* d_in, const int* in_sizes, int n_in,
                              void* d_out, int out_size, void* d_ws, size_t ws_size,
                              hipStream_t stream) {
  (void)in_sizes; (void)n_in; (void)out_size; (void)ws_size;
  const float* A = (const float*)d_in[0];   // (2,64,64) f32
  const float* B = (const float*)d_in[1];   // (2,64,64) f32
  const float* w = (const float*)d_in[2];   // (72,) f32

  float* ws  = (float*)d_ws;
  float* LAr = ws;            float* LAi = ws +   4096;
  float* LBr = ws + 2*4096;   float* LBi = ws + 3*4096;
  float* G   = ws + 4*4096;   // 16 planes of 4096 (GXr,GXi,GYr,GYi per combo)
  float* E   = ws + 20*4096;  // 8 planes of 4096 (Er,Ei per combo)

  size_t shmemSand = (size_t)6 * PLANE * sizeof(float);   // ~102 KB (< 320 KB WGP LDS)
  size_t shmemE    = (size_t)4 * 4096 * sizeof(float);    // 64 KB

  build_L_kernel<<<1, 64, 0, stream>>>(w, LAr, LAi, LBr, LBi);
  phaseA_kernel<<<8, 256, shmemSand, stream>>>(A, B, LAr, LAi, LBr, LBi, G);
  E_kernel<<<4, 256, shmemE, stream>>>(G, E);
  phaseC_kernel<<<4, 256, shmemSand, stream>>>(E, LAr, LAi, (float*)d_out);
}